// Scatter_50757923504892
// MI455X (gfx1250) — compile-verified
//
#include <hip/hip_runtime.h>
#include <hip/hip_bf16.h>
#include <stdint.h>

// Segment softmax (scatter op='softmax') for MI455X / gfx1250.
//
// src [50000,48] f32, index [2,1.6M] i32 (row0=u gather, row1=v scatter),
// out [1.6M,48] f32 = exp(src[u]-segmax[v]) / segsum[v].
//
// Roofline: mandatory HBM traffic is the 307MB output write (~13us @
// 23.3TB/s). src (9.6MB), index (12.8MB), segmax/denom workspace (19.2MB) are
// all L2-resident (192MB L2). No dense matmul -> WMMA inapplicable; gfx1250
// paths used: global_load_async_to_lds_b128 + s_wait_asynccnt for index
// staging, L2-native global_atomic_max_u32 / global_atomic_add_f32, ds_load
// broadcast reads, wave32 blocks, non-temporal b128 output stores.
// VALU kept under the bandwidth floor with v_exp_f32 (__expf) and
// v_rcp_f32 + mul instead of libm expf / IEEE division.

#define N_NODES 50000
#define N_EDGES 1600000
#define D_FEAT  48
#define CHUNKS  (D_FEAT / 4)        // 12 float4 chunks per feature row
#define EPB     64                  // edges per block (1.6M % 64 == 0)
#define THREADS 256                 // 8 wave32
#define WORK    (EPB * CHUNKS)      // 768 work items per block
#define ITERS   (WORK / THREADS)    // 3

typedef __attribute__((ext_vector_type(4))) float f32x4;

// Order-preserving bijection f32 -> u32 so unsigned atomicMax == float max (exact).
__device__ __forceinline__ unsigned int f32_to_ord(float x) {
  unsigned int b = __float_as_uint(x);
  return (b & 0x80000000u) ? ~b : (b | 0x80000000u);
}
__device__ __forceinline__ float ord_to_f32(unsigned int o) {
  return __uint_as_float((o & 0x80000000u) ? (o ^ 0x80000000u) : ~o);
}

// Fast exp: v_mul (x*log2e) + v_exp_f32. ~1e-6 rel error; used identically in
// passes 2 and 3 so numerator and denominator are consistent.
__device__ __forceinline__ float fast_exp(float x) { return __expf(x); }

// Stage this block's 64 u-indices + 64 v-indices (512B) into LDS with one
// wave of async b128 global->LDS copies. Lanes 0..15 fetch the u row,
// lanes 16..31 the v row. Issuing wave drains its ASYNCcnt; barrier
// publishes LDS to the other 7 waves.
__device__ __forceinline__ void stage_indices(const int* __restrict__ idx,
                                              int e0, int* s_idx) {
  if (threadIdx.x < 32) {
    const unsigned lane  = threadIdx.x;
    const unsigned which = lane >> 4;                  // 0: u row, 1: v row
    const unsigned l     = lane & 15u;
    const unsigned goff  = which * (unsigned)(N_EDGES * 4) +
                           (unsigned)e0 * 4u + l * 16u;   // byte offset vs idx base
    const unsigned lds   = (unsigned)(uintptr_t)s_idx + lane * 16u; // LDS byte addr
    asm volatile("global_load_async_to_lds_b128 %0, %1, %2"
                 :: "v"(lds), "v"(goff), "s"(idx) : "memory");
    asm volatile("s_wait_asynccnt 0x0" ::: "memory");
  }
  __syncthreads();
}

// Pass 0: zero workspace (segmax encodings + denom). ord==0 loses to every
// real float's encoding, and 0u is also +0.0f for the denom accumulator.
__global__ __launch_bounds__(THREADS) void ws_init_kernel(uint4* __restrict__ ws, int n4) {
  int i = blockIdx.x * blockDim.x + threadIdx.x;
  if (i < n4) ws[i] = make_uint4(0u, 0u, 0u, 0u);
}

// Pass 1: segmax[v,d] = max over edges of src[u,d] (exact float max via u32 atomics)
__global__ __launch_bounds__(THREADS) void seg_max_kernel(
    const float4* __restrict__ src4, const int* __restrict__ idx,
    unsigned int* __restrict__ segmax) {
  __shared__ __align__(16) int s_idx[2 * EPB];
  const int e0 = blockIdx.x * EPB;
  stage_indices(idx, e0, s_idx);
#pragma unroll
  for (int k = 0; k < ITERS; ++k) {
    int w  = k * THREADS + threadIdx.x;
    int el = w / CHUNKS;
    int c  = w - el * CHUNKS;
    int u  = s_idx[el];            // ds_load broadcast (12 lanes same addr)
    int v  = s_idx[EPB + el];
    float4 x = src4[u * CHUNKS + c];   // 192B coalesced burst, L2-resident
    unsigned int* p = segmax + v * D_FEAT + c * 4;
    atomicMax(p + 0, f32_to_ord(x.x));
    atomicMax(p + 1, f32_to_ord(x.y));
    atomicMax(p + 2, f32_to_ord(x.z));
    atomicMax(p + 3, f32_to_ord(x.w));
  }
}

// Pass 2: denom[v,d] += exp(src[u,d] - segmax[v,d]) (f32 atomic adds into L2)
__global__ __launch_bounds__(THREADS) void seg_sum_kernel(
    const float4* __restrict__ src4, const int* __restrict__ idx,
    const uint4* __restrict__ segmax4, float* __restrict__ denom) {
  __shared__ __align__(16) int s_idx[2 * EPB];
  const int e0 = blockIdx.x * EPB;
  stage_indices(idx, e0, s_idx);
#pragma unroll
  for (int k = 0; k < ITERS; ++k) {
    int w  = k * THREADS + threadIdx.x;
    int el = w / CHUNKS;
    int c  = w - el * CHUNKS;
    int u  = s_idx[el];
    int v  = s_idx[EPB + el];
    float4 x = src4[u * CHUNKS + c];
    uint4  m = segmax4[v * CHUNKS + c];
    float* p = denom + v * D_FEAT + c * 4;
    atomicAdd(p + 0, fast_exp(x.x - ord_to_f32(m.x)));   // global_atomic_add_f32
    atomicAdd(p + 1, fast_exp(x.y - ord_to_f32(m.y)));
    atomicAdd(p + 2, fast_exp(x.z - ord_to_f32(m.z)));
    atomicAdd(p + 3, fast_exp(x.w - ord_to_f32(m.w)));
  }
}

// Pass 3: out[e,d] = exp(src[u,d] - segmax[v,d]) * rcp(denom[v,d]).
// Recomputes exp (cheap: v_exp_f32) instead of spilling a 307MB intermediate;
// v_rcp_f32 + mul instead of IEEE division (denom >= 1, well-conditioned).
// Output stored non-temporally so the write-once stream doesn't evict the
// L2-resident tables. Empty segments are never referenced by v.
__global__ __launch_bounds__(THREADS) void seg_out_kernel(
    const float4* __restrict__ src4, const int* __restrict__ idx,
    const uint4* __restrict__ segmax4, const float4* __restrict__ denom4,
    float* __restrict__ out) {
  __shared__ __align__(16) int s_idx[2 * EPB];
  const int e0 = blockIdx.x * EPB;
  stage_indices(idx, e0, s_idx);
#pragma unroll
  for (int k = 0; k < ITERS; ++k) {
    int w  = k * THREADS + threadIdx.x;
    int el = w / CHUNKS;
    int c  = w - el * CHUNKS;
    int u  = s_idx[el];
    int v  = s_idx[EPB + el];
    float4 x = src4[u * CHUNKS + c];
    uint4  m = segmax4[v * CHUNKS + c];
    float4 d = denom4[v * CHUNKS + c];
    f32x4 o;
    o.x = fast_exp(x.x - ord_to_f32(m.x)) * __builtin_amdgcn_rcpf(d.x);
    o.y = fast_exp(x.y - ord_to_f32(m.y)) * __builtin_amdgcn_rcpf(d.y);
    o.z = fast_exp(x.z - ord_to_f32(m.z)) * __builtin_amdgcn_rcpf(d.z);
    o.w = fast_exp(x.w - ord_to_f32(m.w)) * __builtin_amdgcn_rcpf(d.w);
    // Global work index = blockIdx.x*768 + w: perfectly contiguous NT b128 stores.
    __builtin_nontemporal_store(o, (f32x4*)(out) + (e0 * CHUNKS + w));
  }
}

extern "C" void kernel_launch(void* const* d_in, const int* in_sizes, int n_in,
                              void* d_out, int out_size, void* d_ws, size_t ws_size,
                              hipStream_t stream) {
  const float* src = (const float*)d_in[0];   // [N_NODES, D_FEAT] f32
  const int*   idx = (const int*)d_in[1];     // [2, N_EDGES] i32
  float*       out = (float*)d_out;           // [N_EDGES, D_FEAT] f32

  // Workspace layout (19.2 MB total, L2-resident):
  //   [0 .. 9.6MB)   segmax: N_NODES*D_FEAT u32 (order-encoded floats)
  //   [9.6 .. 19.2)  denom : N_NODES*D_FEAT f32
  unsigned int* segmax = (unsigned int*)d_ws;
  float* denom = (float*)((char*)d_ws + (size_t)N_NODES * D_FEAT * sizeof(unsigned int));
  (void)in_sizes; (void)n_in; (void)out_size; (void)ws_size;

  const int n4 = (2 * N_NODES * D_FEAT) / 4;            // 1.2M uint4 inits
  ws_init_kernel<<<(n4 + THREADS - 1) / THREADS, THREADS, 0, stream>>>((uint4*)d_ws, n4);

  const int blocks = N_EDGES / EPB;                     // 25000 blocks, exact
  seg_max_kernel<<<blocks, THREADS, 0, stream>>>((const float4*)src, idx, segmax);
  seg_sum_kernel<<<blocks, THREADS, 0, stream>>>((const float4*)src, idx,
                                                 (const uint4*)segmax, denom);
  seg_out_kernel<<<blocks, THREADS, 0, stream>>>((const float4*)src, idx,
                                                 (const uint4*)segmax,
                                                 (const float4*)denom, out);
}